// WeightedRankPairwiseLoss_19971597927131
// MI455X (gfx1250) — compile-verified
//
#include <hip/hip_runtime.h>

#define TPB 256
#define MARGINF 1.0f
#define DEPTH 4              // async pipeline depth (LDS buffers)

#ifndef __has_builtin
#define __has_builtin(x) 0
#endif

// Hard compiler barrier + exact CDNA5 wait instruction (ASYNCcnt tracks
// GLOBAL_LOAD_ASYNC_TO_LDS_*; see CDNA5 ISA).
#define WAIT_ASYNCCNT(n) asm volatile("s_wait_asynccnt %0" :: "n"(n) : "memory")

// Vector type matching the builtin's expected pointee (per hipcc diagnostic:
// "__attribute__((__vector_size__(4 * sizeof(int)))) int")
typedef int v4i_gcc __attribute__((vector_size(16)));
typedef __attribute__((address_space(1))) v4i_gcc* glob_v4i_ptr;
typedef __attribute__((address_space(3))) v4i_gcc* lds_v4i_ptr;

// Issue one per-lane 16B async copy global -> LDS (GLOBAL_LOAD_ASYNC_TO_LDS_B128,
// CDNA5 ISA). Each lane writes only its own LDS slot, so a
// wave only ever depends on its own ASYNCcnt (no cross-wave barriers needed).
__device__ __forceinline__ void async_g2l_b128(const float* gptr, float* lptr) {
#if __has_builtin(__builtin_amdgcn_global_load_async_to_lds_b128)
  __builtin_amdgcn_global_load_async_to_lds_b128(
      (glob_v4i_ptr)gptr,
      (lds_v4i_ptr)lptr,
      /*imm offset=*/0, /*cpol=*/0);
#else
  unsigned lds_off =
      (unsigned)(unsigned long long)(__attribute__((address_space(3))) char*)lptr;
  unsigned long long ga = (unsigned long long)gptr;
  asm volatile("global_load_async_to_lds_b128 %0, %1, off"
               :: "v"(lds_off), "v"(ga) : "memory");
#endif
}

__device__ __forceinline__ float block_reduce_sum(float v, float* buf, int tid) {
  __syncthreads();
  buf[tid] = v;
  __syncthreads();
  #pragma unroll
  for (int s = TPB >> 1; s > 0; s >>= 1) {
    if (tid < s) buf[tid] += buf[tid + s];
    __syncthreads();
  }
  float r = buf[0];
  __syncthreads();
  return r;
}

__global__ void __launch_bounds__(TPB)
wrp_row_kernel(const float* __restrict__ scores,
               const int* __restrict__ targets,
               float* __restrict__ row_loss,
               int C) {
  const int row = blockIdx.x;
  const int tid = threadIdx.x;

  __shared__ __align__(16) float lbuf[DEPTH][TPB * 4];   // 16 KB, 4-deep pipeline
  __shared__ float redf[TPB];
  __shared__ float sh_gt;
  __shared__ int sh_t;
  __shared__ unsigned sh_npos, sh_cgt, sh_ceq;

  const float* rowp = scores + (size_t)row * (size_t)C;

  if (tid == 0) {
    int t = targets[row];
    sh_t = t;
    sh_gt = rowp[t];
    sh_npos = 0u; sh_cgt = 0u; sh_ceq = 0u;
  }
  __syncthreads();
  const float gt  = sh_gt;
  const int   tgt = sh_t;

  unsigned cgt = 0u, ceq = 0u, np = 0u;
  float hs = 0.0f;

  const int nVec   = C >> 2;               // number of float4s in the row
  const int nTiles = nVec / TPB;           // full async tiles (TPB*16 B each)
  const int remVec = nVec - nTiles * TPB;  // leftover float4s

  // ---- prologue: fill pipeline with up to DEPTH-1 tiles ----
  const int pro = (nTiles < (DEPTH - 1)) ? nTiles : (DEPTH - 1);
  for (int p = 0; p < pro; ++p) {
    async_g2l_b128(rowp + ((size_t)p * TPB + tid) * 4, &lbuf[p][tid * 4]);
  }

  // ---- steady-state: issue tile t+DEPTH-1, wait for tile t, consume ----
  for (int tile = 0; tile < nTiles; ++tile) {
    const int nxt = tile + (DEPTH - 1);
    if (nxt < nTiles) {
      // buf[nxt % DEPTH] was last read at iteration tile-1: safe to overwrite
      async_g2l_b128(rowp + ((size_t)nxt * TPB + tid) * 4,
                     &lbuf[nxt & (DEPTH - 1)][tid * 4]);
      WAIT_ASYNCCNT(DEPTH - 1);   // oldest in-flight tile (== `tile`) landed
    } else {
      WAIT_ASYNCCNT(0);           // tail: drain (last few tiles only)
    }
    float4 v = *reinterpret_cast<const float4*>(&lbuf[tile & (DEPTH - 1)][tid * 4]);
    const int j0 = (tile * TPB + tid) * 4;
    const float e[4] = {v.x, v.y, v.z, v.w};
    #pragma unroll
    for (int k = 0; k < 4; ++k) {
      const float s = e[k];
      const int   j = j0 + k;
      cgt += (s > gt) ? 1u : 0u;
      ceq += ((s == gt) && (j < tgt)) ? 1u : 0u;
      const float l = MARGINF + s - gt;
      if (l > 0.0f) { hs += l; np += 1u; }
    }
  }
  // ---- remainder float4s (direct global b128 loads) ----
  if (tid < remVec) {
    const int jv = nTiles * TPB + tid;
    float4 v = *reinterpret_cast<const float4*>(rowp + (size_t)jv * 4);
    const int j0 = jv * 4;
    const float e[4] = {v.x, v.y, v.z, v.w};
    #pragma unroll
    for (int k = 0; k < 4; ++k) {
      const float s = e[k];
      const int   j = j0 + k;
      cgt += (s > gt) ? 1u : 0u;
      ceq += ((s == gt) && (j < tgt)) ? 1u : 0u;
      const float l = MARGINF + s - gt;
      if (l > 0.0f) { hs += l; np += 1u; }
    }
  }
  // ---- scalar tail (C % 4) ----
  const int tail = C & 3;
  if (tid < tail) {
    const int j = (C & ~3) + tid;
    const float s = rowp[j];
    cgt += (s > gt) ? 1u : 0u;
    ceq += ((s == gt) && (j < tgt)) ? 1u : 0u;
    const float l = MARGINF + s - gt;
    if (l > 0.0f) { hs += l; np += 1u; }
  }

  // ---- block reductions (deterministic) ----
  float hinge = block_reduce_sum(hs, redf, tid);
  atomicAdd(&sh_npos, np);
  atomicAdd(&sh_cgt, cgt);
  atomicAdd(&sh_ceq, ceq);
  __syncthreads();
  const unsigned npos_u = sh_npos - 1u;          // drop j==t (its l == MARGIN > 0)
  hinge -= MARGINF;                              // drop j==t contribution
  const int rank = 1 + (int)sh_cgt + (int)sh_ceq;

  // H(rank) = sum_{i=1..rank} 1/i, parallel across the block
  float h = 0.0f;
  for (int i = tid + 1; i <= rank; i += TPB) h += 1.0f / (float)i;
  const float H = block_reduce_sum(h, redf, tid);

  if (tid == 0) {
    row_loss[row] = H / ((float)npos_u + 1e-7f) * hinge;
  }
}

__global__ void __launch_bounds__(TPB)
wrp_finalize_kernel(const float* __restrict__ row_loss,
                    float* __restrict__ out, int B) {
  __shared__ float buf[TPB];
  const int tid = threadIdx.x;
  float s = 0.0f;
  for (int i = tid; i < B; i += TPB) s += row_loss[i];   // fixed order
  buf[tid] = s;
  __syncthreads();
  #pragma unroll
  for (int off = TPB >> 1; off > 0; off >>= 1) {
    if (tid < off) buf[tid] += buf[tid + off];
    __syncthreads();
  }
  if (tid == 0) out[0] = buf[0] / (float)B;
}

extern "C" void kernel_launch(void* const* d_in, const int* in_sizes, int n_in,
                              void* d_out, int out_size, void* d_ws, size_t ws_size,
                              hipStream_t stream) {
  const float* scores  = (const float*)d_in[0];
  const int*   targets = (const int*)d_in[1];
  const int B = in_sizes[1];
  const int C = in_sizes[0] / B;
  float* row_loss = (float*)d_ws;   // B floats of scratch

  wrp_row_kernel<<<B, TPB, 0, stream>>>(scores, targets, row_loss, C);
  wrp_finalize_kernel<<<1, TPB, 0, stream>>>(row_loss, (float*)d_out, B);
}